// LocalFeatureAggregation_541165879358
// MI455X (gfx1250) — compile-verified
//
#include <hip/hip_runtime.h>
#include <hip/hip_bf16.h>

#define NPTS 8192
#define KNN 16
#define CHUNK 2048
#define ROWP 66   // padded LDS row stride (floats) to avoid bank conflicts

typedef __attribute__((ext_vector_type(2))) float v2f;
typedef __attribute__((ext_vector_type(8))) float v8f;

__device__ __forceinline__ float leakyf(float x, float s) { return x >= 0.f ? x : s * x; }

// ---------------------------------------------------------------------------
// KNN: one thread per query point; batch coords staged through LDS in chunks.
// Register top-16 insertion (stable, strict <) matches jax top_k ordering.
// d2 = n2_i + n2_j - 2*dot  (reference formula).
// ---------------------------------------------------------------------------
__global__ void knn_kernel(const float* __restrict__ coords,
                           int* __restrict__ idx_out, float* __restrict__ dist_out) {
  __shared__ float sc[CHUNK * 3];
  const int b = blockIdx.y;
  const int qi = blockIdx.x * blockDim.x + threadIdx.x;   // 0..8191
  const float* cb = coords + (size_t)b * NPTS * 3;
  const float qx = cb[qi * 3 + 0], qy = cb[qi * 3 + 1], qz = cb[qi * 3 + 2];
  const float qn2 = qx * qx + qy * qy + qz * qz;
  float d[KNN]; int id[KNN];
#pragma unroll
  for (int s = 0; s < KNN; ++s) { d[s] = 3.4e38f; id[s] = 0; }

  for (int ch = 0; ch < NPTS; ch += CHUNK) {
    for (int t = threadIdx.x; t < CHUNK * 3; t += blockDim.x) sc[t] = cb[ch * 3 + t];
    __syncthreads();
    for (int j = 0; j < CHUNK; ++j) {
      const float cx = sc[j * 3 + 0], cy = sc[j * 3 + 1], cz = sc[j * 3 + 2];
      const float nn2 = cx * cx + cy * cy + cz * cz;
      const float dot = qx * cx + qy * cy + qz * cz;
      const float d2 = qn2 + nn2 - 2.f * dot;
      if (d2 < d[KNN - 1]) {
        d[KNN - 1] = d2; id[KNN - 1] = ch + j;
#pragma unroll
        for (int s = KNN - 1; s > 0; --s) {
          if (d[s] < d[s - 1]) {
            float td = d[s]; d[s] = d[s - 1]; d[s - 1] = td;
            int ti = id[s]; id[s] = id[s - 1]; id[s - 1] = ti;
          }
        }
      }
    }
    __syncthreads();
  }
  const int base = (b * NPTS + qi) * KNN;
#pragma unroll
  for (int s = 0; s < KNN; ++s) { idx_out[base + s] = id[s]; dist_out[base + s] = d[s]; }
}

// ---------------------------------------------------------------------------
// mlp1: x1(b,n,32) = leaky_relu(w(32x32) @ features(b,32,n) + b, 0.2)
// ---------------------------------------------------------------------------
__global__ void mlp1_kernel(const float* __restrict__ feats, const float* __restrict__ w,
                            const float* __restrict__ bias, float* __restrict__ x1) {
  const int bn = blockIdx.x * blockDim.x + threadIdx.x;   // 0..16383
  if (bn >= 2 * NPTS) return;
  const int b = bn >> 13, n = bn & (NPTS - 1);
  const float* fb = feats + ((size_t)b * 32) * NPTS + n;
  float f[32];
#pragma unroll
  for (int c = 0; c < 32; ++c) f[c] = fb[(size_t)c * NPTS];
  for (int o = 0; o < 32; ++o) {
    float a = bias[o];
#pragma unroll
    for (int c = 0; c < 32; ++c) a = fmaf(w[o * 32 + c], f[c], a);
    x1[(size_t)bn * 32 + o] = leakyf(a, 0.2f);
  }
}

// ---------------------------------------------------------------------------
// LSE conv: per (b,n,k) build 10-ch cat = [ext, nbr, ext-nbr, dist] implicitly,
// enc_raw(b,n,k,32) = w(32x10) @ cat + b
// ---------------------------------------------------------------------------
__global__ void lse_conv_kernel(const float* __restrict__ coords, const int* __restrict__ idx,
                                const float* __restrict__ dist, const float* __restrict__ w,
                                const float* __restrict__ bias, float* __restrict__ enc_raw) {
  const int t = blockIdx.x * blockDim.x + threadIdx.x;    // 0..262143
  if (t >= 2 * NPTS * KNN) return;
  const int bn = t >> 4;
  const int b = bn >> 13, n = bn & (NPTS - 1);
  const float* cb = coords + (size_t)b * NPTS * 3;
  const float ex = cb[n * 3 + 0], ey = cb[n * 3 + 1], ez = cb[n * 3 + 2];
  const int j = idx[t];
  const float nx = cb[j * 3 + 0], ny = cb[j * 3 + 1], nz = cb[j * 3 + 2];
  const float cat[10] = {ex, ey, ez, nx, ny, nz, ex - nx, ey - ny, ez - nz, dist[t]};
  float* out = enc_raw + (size_t)t * 32;
  for (int o = 0; o < 32; ++o) {
    float a = bias[o];
#pragma unroll
    for (int c = 0; c < 10; ++c) a = fmaf(w[o * 10 + c], cat[c], a);
    out[o] = a;
  }
}

// ---------------------------------------------------------------------------
// BN stats: one block per channel, deterministic serial+tree reduction.
// src layout (i, c) contiguous in c. stats[c]=mean, stats[C+c]=rsqrt(var+eps).
// ---------------------------------------------------------------------------
__global__ void stats_kernel(const float* __restrict__ src, int C, int count,
                             float* __restrict__ stats) {
  __shared__ float ssum[256], ssq[256];
  const int c = blockIdx.x;
  float s = 0.f, q = 0.f;
  for (int i = threadIdx.x; i < count; i += 256) {
    const float v = src[(size_t)i * C + c];
    s += v; q = fmaf(v, v, q);
  }
  ssum[threadIdx.x] = s; ssq[threadIdx.x] = q;
  __syncthreads();
  for (int off = 128; off > 0; off >>= 1) {
    if (threadIdx.x < off) {
      ssum[threadIdx.x] += ssum[threadIdx.x + off];
      ssq[threadIdx.x] += ssq[threadIdx.x + off];
    }
    __syncthreads();
  }
  if (threadIdx.x == 0) {
    const float mu = ssum[0] / (float)count;
    const float var = ssq[0] / (float)count - mu * mu;
    stats[c] = mu;
    stats[C + c] = rsqrtf(var + 1e-6f);
  }
}

// ---------------------------------------------------------------------------
// Elementwise BN + relu
// ---------------------------------------------------------------------------
__global__ void bn_relu_kernel(const float* __restrict__ src, const float* __restrict__ stats,
                               const float* __restrict__ g, const float* __restrict__ be,
                               float* __restrict__ dst, int C, int total) {
  const int i = blockIdx.x * blockDim.x + threadIdx.x;
  if (i >= total) return;
  const int c = i % C;
  const float y = g[c] * (src[i] - stats[c]) * stats[C + c] + be[c];
  dst[i] = fmaxf(y, 0.f);
}

// ---------------------------------------------------------------------------
// Fused attention-score (WMMA f32 16x16x4) + softmax(K) + weighted pool.
// One wave per point n: builds x (64ch x 16k) in LDS (BN+relu of enc_raw for
// c<32, broadcast feats for c>=32), computes S = Wscore(64x64) @ x via
// 4 M-tiles x 16 K-steps of V_WMMA_F32_16X16X4_F32, softmax over the 16
// neighbors (16-lane shfl reductions, wave32), pooled[c] = sum_k P[c,k]*x[c,k].
// Operand layouts per ISA: A/B: lane%16 = M/N, K = vgpr + 2*(lane/16);
// D: M = vgpr + 8*(lane/16), N = lane%16.
// ---------------------------------------------------------------------------
__global__ void __launch_bounds__(128) score_pool_kernel(
    const float* __restrict__ enc_raw, const float* __restrict__ stats,
    const float* __restrict__ g, const float* __restrict__ be,
    const float* __restrict__ xsrc, const float* __restrict__ wscore,
    float* __restrict__ pooled) {
  __shared__ float xl[4][16 * ROWP];
  const int wave = threadIdx.x >> 5;
  const int lane = threadIdx.x & 31;
  const int bn = blockIdx.x * 4 + wave;                   // 0..16383
  float* xw = &xl[wave][0];

  const float* er = enc_raw + (size_t)bn * KNN * 32;
  const float* xs = xsrc + (size_t)bn * 32;
  for (int i = lane; i < 16 * 64; i += 32) {
    const int k = i >> 6, c = i & 63;
    float v;
    if (c < 32) {
      const float r = er[k * 32 + c];
      v = fmaxf(g[c] * (r - stats[c]) * stats[32 + c] + be[c], 0.f);
    } else {
      v = xs[c - 32];
    }
    xw[k * ROWP + c] = v;
  }
  __syncthreads();

  const int half = lane >> 4, lq = lane & 15;
  for (int m = 0; m < 4; ++m) {
    v8f acc = {0.f, 0.f, 0.f, 0.f, 0.f, 0.f, 0.f, 0.f};
    const float* wrow = wscore + (m * 16 + lq) * 64 + 2 * half;
    const float* xcol = xw + lq * ROWP + 2 * half;
#pragma unroll
    for (int ks = 0; ks < 16; ++ks) {
      v2f a;  a.x = wrow[ks * 4]; a.y = wrow[ks * 4 + 1];
      v2f bb; bb.x = xcol[ks * 4]; bb.y = xcol[ks * 4 + 1];
      acc = __builtin_amdgcn_wmma_f32_16x16x4_f32(false, a, false, bb,
                                                  (short)0, acc, false, false);
    }
#pragma unroll
    for (int v = 0; v < 8; ++v) {
      const float s = acc[v];
      float mx = s;
      for (int off = 8; off > 0; off >>= 1) mx = fmaxf(mx, __shfl_xor(mx, off, 16));
      const float e = __expf(s - mx);
      float sum = e;
      for (int off = 8; off > 0; off >>= 1) sum += __shfl_xor(sum, off, 16);
      const int mch = m * 16 + v + 8 * half;
      float pv = (e / sum) * xw[lq * ROWP + mch];
      for (int off = 8; off > 0; off >>= 1) pv += __shfl_xor(pv, off, 16);
      if (lq == 0) pooled[(size_t)bn * 64 + mch] = pv;
    }
  }
}

// ---------------------------------------------------------------------------
// 1x1 conv over row-major input (bn, C): out(bn, O) = w(OxC) @ in + bias
// ---------------------------------------------------------------------------
__global__ void conv_row_kernel(const float* __restrict__ in, const float* __restrict__ w,
                                const float* __restrict__ bias, float* __restrict__ out,
                                int C, int O) {
  const int t = blockIdx.x * blockDim.x + threadIdx.x;
  if (t >= 2 * NPTS * O) return;
  const int o = t % O;
  const int bn = t / O;
  const float* row = in + (size_t)bn * C;
  float a = bias[o];
  for (int c = 0; c < C; ++c) a = fmaf(w[o * C + c], row[c], a);
  out[t] = a;
}

// ---------------------------------------------------------------------------
// 1x1 conv over channel-first input (b, C, n): for the shortcut conv.
// ---------------------------------------------------------------------------
__global__ void conv_chfirst_kernel(const float* __restrict__ in, const float* __restrict__ w,
                                    const float* __restrict__ bias, float* __restrict__ out,
                                    int C, int O) {
  const int t = blockIdx.x * blockDim.x + threadIdx.x;
  if (t >= 2 * NPTS * O) return;
  const int o = t % O;
  const int bn = t / O;
  const int b = bn >> 13, n = bn & (NPTS - 1);
  const float* fb = in + ((size_t)b * C) * NPTS + n;
  float a = bias[o];
  for (int c = 0; c < C; ++c) a = fmaf(w[o * C + c], fb[(size_t)c * NPTS], a);
  out[(size_t)bn * O + o] = a;
}

// ---------------------------------------------------------------------------
// Final: out(b,128,n) = leaky( w_mlp2(128x64)@x3(bn,64) + b + BN(sc_raw), 0.01)
// ---------------------------------------------------------------------------
__global__ void final_kernel(const float* __restrict__ x3, const float* __restrict__ sc_raw,
                             const float* __restrict__ stats_sc, const float* __restrict__ g,
                             const float* __restrict__ be, const float* __restrict__ w,
                             const float* __restrict__ bias, float* __restrict__ out) {
  const int t = blockIdx.x * blockDim.x + threadIdx.x;    // 2*128*8192
  if (t >= 2 * 128 * NPTS) return;
  const int n = t & (NPTS - 1);
  const int r = t >> 13;
  const int oc = r & 127;
  const int b = r >> 7;
  const int bn = b * NPTS + n;
  const float* row = x3 + (size_t)bn * 64;
  float a = bias[oc];
#pragma unroll
  for (int c = 0; c < 64; ++c) a = fmaf(w[oc * 64 + c], row[c], a);
  const float sc = sc_raw[(size_t)bn * 128 + oc];
  const float scn = g[oc] * (sc - stats_sc[oc]) * stats_sc[128 + oc] + be[oc];
  out[((size_t)b * 128 + oc) * NPTS + n] = leakyf(a + scn, 0.01f);
}

// ---------------------------------------------------------------------------
extern "C" void kernel_launch(void* const* d_in, const int* in_sizes, int n_in,
                              void* d_out, int out_size, void* d_ws, size_t ws_size,
                              hipStream_t stream) {
  (void)in_sizes; (void)n_in; (void)out_size; (void)ws_size;
  const float* coords   = (const float*)d_in[0];
  const float* features = (const float*)d_in[1];
  const float* w_mlp1   = (const float*)d_in[2];
  const float* b_mlp1   = (const float*)d_in[3];
  const float* w_lse1   = (const float*)d_in[4];
  const float* b_lse1   = (const float*)d_in[5];
  const float* g_lse1   = (const float*)d_in[6];
  const float* be_lse1  = (const float*)d_in[7];
  const float* w_score1 = (const float*)d_in[8];
  const float* w_pool1  = (const float*)d_in[9];
  const float* b_pool1  = (const float*)d_in[10];
  const float* g_pool1  = (const float*)d_in[11];
  const float* be_pool1 = (const float*)d_in[12];
  const float* w_lse2   = (const float*)d_in[13];
  const float* b_lse2   = (const float*)d_in[14];
  const float* g_lse2   = (const float*)d_in[15];
  const float* be_lse2  = (const float*)d_in[16];
  const float* w_score2 = (const float*)d_in[17];
  const float* w_pool2  = (const float*)d_in[18];
  const float* b_pool2  = (const float*)d_in[19];
  const float* g_pool2  = (const float*)d_in[20];
  const float* be_pool2 = (const float*)d_in[21];
  const float* w_mlp2   = (const float*)d_in[22];
  const float* b_mlp2   = (const float*)d_in[23];
  const float* w_short  = (const float*)d_in[24];
  const float* b_short  = (const float*)d_in[25];
  const float* g_short  = (const float*)d_in[26];
  const float* be_short = (const float*)d_in[27];

  float* ws = (float*)d_ws;
  size_t off = 0;
  float* dist   = ws + off; off += (size_t)2 * NPTS * KNN;        // 1 MB
  int*   idx    = (int*)(ws + off); off += (size_t)2 * NPTS * KNN;
  float* x1     = ws + off; off += (size_t)2 * NPTS * 32;
  float* enc    = ws + off; off += (size_t)2 * NPTS * KNN * 32;   // 32 MB
  float* pooled = ws + off; off += (size_t)2 * NPTS * 64;
  float* praw1  = ws + off; off += (size_t)2 * NPTS * 32;
  float* x2     = ws + off; off += (size_t)2 * NPTS * 32;
  float* praw2  = ws + off; off += (size_t)2 * NPTS * 64;
  float* x3     = ws + off; off += (size_t)2 * NPTS * 64;
  float* scraw  = ws + off; off += (size_t)2 * NPTS * 128;
  float* st_lse1 = ws + off; off += 64;
  float* st_p1   = ws + off; off += 64;
  float* st_lse2 = ws + off; off += 64;
  float* st_p2   = ws + off; off += 128;
  float* st_sc   = ws + off; off += 256;

  knn_kernel<<<dim3(NPTS / 256, 2), 256, 0, stream>>>(coords, idx, dist);
  mlp1_kernel<<<(2 * NPTS + 255) / 256, 256, 0, stream>>>(features, w_mlp1, b_mlp1, x1);

  // Round 1
  lse_conv_kernel<<<(2 * NPTS * KNN + 255) / 256, 256, 0, stream>>>(coords, idx, dist, w_lse1, b_lse1, enc);
  stats_kernel<<<32, 256, 0, stream>>>(enc, 32, 2 * NPTS * KNN, st_lse1);
  score_pool_kernel<<<(2 * NPTS) / 4, 128, 0, stream>>>(enc, st_lse1, g_lse1, be_lse1, x1, w_score1, pooled);
  conv_row_kernel<<<(2 * NPTS * 32 + 255) / 256, 256, 0, stream>>>(pooled, w_pool1, b_pool1, praw1, 64, 32);
  stats_kernel<<<32, 256, 0, stream>>>(praw1, 32, 2 * NPTS, st_p1);
  bn_relu_kernel<<<(2 * NPTS * 32 + 255) / 256, 256, 0, stream>>>(praw1, st_p1, g_pool1, be_pool1, x2, 32, 2 * NPTS * 32);

  // Round 2
  lse_conv_kernel<<<(2 * NPTS * KNN + 255) / 256, 256, 0, stream>>>(coords, idx, dist, w_lse2, b_lse2, enc);
  stats_kernel<<<32, 256, 0, stream>>>(enc, 32, 2 * NPTS * KNN, st_lse2);
  score_pool_kernel<<<(2 * NPTS) / 4, 128, 0, stream>>>(enc, st_lse2, g_lse2, be_lse2, x2, w_score2, pooled);
  conv_row_kernel<<<(2 * NPTS * 64 + 255) / 256, 256, 0, stream>>>(pooled, w_pool2, b_pool2, praw2, 64, 64);
  stats_kernel<<<64, 256, 0, stream>>>(praw2, 64, 2 * NPTS, st_p2);
  bn_relu_kernel<<<(2 * NPTS * 64 + 255) / 256, 256, 0, stream>>>(praw2, st_p2, g_pool2, be_pool2, x3, 64, 2 * NPTS * 64);

  // Shortcut + final
  conv_chfirst_kernel<<<(2 * NPTS * 128 + 255) / 256, 256, 0, stream>>>(features, w_short, b_short, scraw, 32, 128);
  stats_kernel<<<128, 256, 0, stream>>>(scraw, 128, 2 * NPTS, st_sc);
  final_kernel<<<(2 * 128 * NPTS + 255) / 256, 256, 0, stream>>>(x3, scraw, st_sc, g_short, be_short,
                                                                 w_mlp2, b_mlp2, (float*)d_out);
}